// GConv_50199577756043
// MI455X (gfx1250) — compile-verified
//
#include <hip/hip_runtime.h>

typedef __attribute__((ext_vector_type(2))) float v2f;
typedef __attribute__((ext_vector_type(8))) float v8f;

// ---------------------------------------------------------------------------
// Small utility kernels
// ---------------------------------------------------------------------------
__global__ void zero_f32(float* __restrict__ p, int n) {
    int i = blockIdx.x * blockDim.x + threadIdx.x;
    if (i < n) p[i] = 0.0f;
}

__global__ void count_deg(const int* __restrict__ dst, float* __restrict__ deg, int E) {
    int i = blockIdx.x * blockDim.x + threadIdx.x;
    if (i < E) atomicAdd(&deg[dst[i]], 1.0f);
}

__global__ void finalize_dinv(const float* __restrict__ deg, float* __restrict__ dinv, int n) {
    int i = blockIdx.x * blockDim.x + threadIdx.x;
    if (i < n) dinv[i] = rsqrtf(deg[i] + 1.0f);   // +1 self loop
}

// W1eff[k][c] = W1[k][c] + W1[k+64][c]   (because xr = [x | x])
__global__ void fold_w1(const float* __restrict__ W1, float* __restrict__ W1e) {
    int i = blockIdx.x * blockDim.x + threadIdx.x;
    if (i < 64 * 256) W1e[i] = W1[i] + W1[i + 64 * 256];
}

// z[i,c] = h[i,c] * dinv[i]^2 + bias[c]   (self-loop term + bias, init for scatter)
__global__ void init_self(const float* __restrict__ h, const float* __restrict__ dinv,
                          const float* __restrict__ bias, float* __restrict__ z,
                          int total, int F) {
    int i = blockIdx.x * blockDim.x + threadIdx.x;
    if (i < total) {
        int row = i / F;
        int col = i - row * F;
        float di = dinv[row];
        z[i] = h[i] * di * di + bias[col];
    }
}

// One wave per edge: out[dst] += dinv[src]*dinv[dst] * h[src]  (lane-strided cols)
__global__ void edge_scatter(const float* __restrict__ h, float* __restrict__ out,
                             const int* __restrict__ src, const int* __restrict__ dst,
                             const float* __restrict__ dinv, int E, int F) {
    int e = blockIdx.x * (blockDim.x >> 5) + (threadIdx.x >> 5);
    if (e >= E) return;
    int lane = threadIdx.x & 31;
    int s = src[e], d = dst[e];
    float norm = dinv[s] * dinv[d];
    const float* hs = h + (size_t)s * F;
    float* od = out + (size_t)d * F;
    for (int c = lane; c < F; c += 32)
        atomicAdd(&od[c], norm * hs[c]);
}

// Column-wise sum / sum-of-squares (blockDim.x must == F, coalesced rows)
__global__ void colsum(const float* __restrict__ z, float* __restrict__ csum,
                       float* __restrict__ csq, int n, int F) {
    int c = threadIdx.x;
    float s = 0.0f, q = 0.0f;
    for (int r = blockIdx.x; r < n; r += gridDim.x) {
        float v = z[(size_t)r * F + c];
        s += v;
        q += v * v;
    }
    atomicAdd(&csum[c], s);
    atomicAdd(&csq[c], q);
}

__global__ void bn_finalize(const float* __restrict__ csum, const float* __restrict__ csq,
                            const float* __restrict__ gamma, const float* __restrict__ beta,
                            float* __restrict__ scale, float* __restrict__ shift,
                            int n, int F) {
    int c = blockIdx.x * blockDim.x + threadIdx.x;
    if (c < F) {
        float inv_n = 1.0f / (float)n;
        float mean  = csum[c] * inv_n;
        float var   = csq[c] * inv_n - mean * mean;
        float sc    = gamma[c] * rsqrtf(var + 1e-5f);
        scale[c] = sc;
        shift[c] = beta[c] - mean * sc;
    }
}

__global__ void bn_apply_relu(float* __restrict__ z, const float* __restrict__ scale,
                              const float* __restrict__ shift, int total, int F) {
    int i = blockIdx.x * blockDim.x + threadIdx.x;
    if (i < total) {
        int c = i % F;
        float v = z[i] * scale[c] + shift[c];
        z[i] = v > 0.0f ? v : 0.0f;
    }
}

// ---------------------------------------------------------------------------
// fp32 WMMA GEMM:  C[M,Nout] = A[M,K] @ W[K,Nout]
// One wave computes a 16(M) x 64(N) tile with 4 v8f accumulators,
// K-loop in steps of 4 via V_WMMA_F32_16X16X4_F32.
// Requires: M % 16 == 0, K % 4 == 0, Nout % 64 == 0.
// Layouts per CDNA5 ISA 7.12.2:
//   A 16x4 : lane m = lane&15, VGPR{0,1} hold K = kb, kb+1 with kb = 2*(lane>>4)
//   B 4x16 : lane n = lane&15, VGPR{0,1} hold K = kb, kb+1 (mirrored)
//   C 16x16: VGPR r holds M = r + 8*(lane>>4), N = lane&15
// ---------------------------------------------------------------------------
__global__ __launch_bounds__(256) void wmma_gemm_f32(const float* __restrict__ A,
                                                     const float* __restrict__ W,
                                                     float* __restrict__ C,
                                                     int M, int K, int Nout, int nTileN64) {
    int wave  = (blockIdx.x << 3) + (threadIdx.x >> 5);
    int total = (M >> 4) * nTileN64;
    if (wave >= total) return;                 // uniform per wave -> EXEC stays full

    int mi = wave / nTileN64;
    int nj = wave - mi * nTileN64;
    int lane = threadIdx.x & 31;
    int m    = (mi << 4) + (lane & 15);        // A row for this lane
    int kb   = (lane >> 4) << 1;               // 0 or 2: K sub-offset for this lane half
    int n0   = nj << 6;                        // first of 64 output columns
    int nl   = n0 + (lane & 15);               // B/C column for this lane

    v8f acc0 = {}, acc1 = {}, acc2 = {}, acc3 = {};
    const float* arow = A + (size_t)m * K;

    for (int k0 = 0; k0 < K; k0 += 4) {
        v2f a;
        a.x = arow[k0 + kb];
        a.y = arow[k0 + kb + 1];

        const float* w0 = W + (size_t)(k0 + kb) * Nout;
        const float* w1 = w0 + Nout;

        v2f b0, b1, b2, b3;
        b0.x = w0[nl];       b0.y = w1[nl];
        b1.x = w0[nl + 16];  b1.y = w1[nl + 16];
        b2.x = w0[nl + 32];  b2.y = w1[nl + 32];
        b3.x = w0[nl + 48];  b3.y = w1[nl + 48];

        acc0 = __builtin_amdgcn_wmma_f32_16x16x4_f32(false, a, false, b0, (short)0, acc0, false, false);
        acc1 = __builtin_amdgcn_wmma_f32_16x16x4_f32(false, a, false, b1, (short)0, acc1, false, false);
        acc2 = __builtin_amdgcn_wmma_f32_16x16x4_f32(false, a, false, b2, (short)0, acc2, false, false);
        acc3 = __builtin_amdgcn_wmma_f32_16x16x4_f32(false, a, false, b3, (short)0, acc3, false, false);
    }

    int mrow0 = (mi << 4) + ((lane >> 4) << 3);   // +8 for upper lane half
    for (int r = 0; r < 8; ++r) {
        size_t base = (size_t)(mrow0 + r) * Nout + nl;
        C[base]      = acc0[r];
        C[base + 16] = acc1[r];
        C[base + 32] = acc2[r];
        C[base + 48] = acc3[r];
    }
}

// ---------------------------------------------------------------------------
// Launch
// ---------------------------------------------------------------------------
extern "C" void kernel_launch(void* const* d_in, const int* in_sizes, int n_in,
                              void* d_out, int out_size, void* d_ws, size_t ws_size,
                              hipStream_t stream) {
    const float* x     = (const float*)d_in[0];   // [N,64]
    const int*   ei    = (const int*)  d_in[1];   // [2,E]
    const float* W1    = (const float*)d_in[2];   // [128,256]
    const float* b1    = (const float*)d_in[3];   // [256]
    const float* gamma = (const float*)d_in[4];   // [256]
    const float* beta  = (const float*)d_in[5];   // [256]
    const float* W2    = (const float*)d_in[6];   // [256,128]
    const float* b2    = (const float*)d_in[7];   // [128]
    float* out = (float*)d_out;                   // [N,128]

    const int N = in_sizes[0] / 64;
    const int E = in_sizes[1] / 2;
    const int F1 = 256, F2 = 128;

    const int* src = ei;
    const int* dst = ei + E;

    // Workspace layout (floats)
    float* ws    = (float*)d_ws;
    float* deg   = ws;                         // N
    float* dinv  = deg + N;                    // N
    float* w1e   = dinv + N;                   // 64*256
    float* csum  = w1e + 64 * 256;             // 256
    float* csq   = csum + 256;                 // 256
    float* scale = csq + 256;                  // 256
    float* shift = scale + 256;                // 256
    float* h1    = shift + 256;                // N*256
    float* z     = h1 + (size_t)N * F1;        // N*256
    float* h2    = z + (size_t)N * F1;         // N*128

    const int T = 256;
    auto cdiv = [](int a, int b) { return (a + b - 1) / b; };

    // 1) degree + dinv
    zero_f32<<<cdiv(N, T), T, 0, stream>>>(deg, N);
    count_deg<<<cdiv(E, T), T, 0, stream>>>(dst, deg, E);
    finalize_dinv<<<cdiv(N, T), T, 0, stream>>>(deg, dinv, N);

    // 2) fold W1 (xr = [x|x])
    fold_w1<<<cdiv(64 * 256, T), T, 0, stream>>>(W1, w1e);

    // 3) h1 = x @ W1eff   [N,64]x[64,256]
    {
        int nT64 = F1 / 64;
        int waves = (N / 16) * nT64;
        wmma_gemm_f32<<<cdiv(waves, 8), T, 0, stream>>>(x, w1e, h1, N, 64, F1, nT64);
    }

    // 4) z = h1/deg + b1, then scatter edges
    init_self<<<cdiv(N * F1, T), T, 0, stream>>>(h1, dinv, b1, z, N * F1, F1);
    edge_scatter<<<cdiv(E, 8), T, 0, stream>>>(h1, z, src, dst, dinv, E, F1);

    // 5) BatchNorm (training stats) + ReLU, in place on z
    zero_f32<<<cdiv(2 * F1, T), T, 0, stream>>>(csum, 2 * F1);   // csum & csq contiguous
    colsum<<<256, F1, 0, stream>>>(z, csum, csq, N, F1);
    bn_finalize<<<cdiv(F1, T), T, 0, stream>>>(csum, csq, gamma, beta, scale, shift, N, F1);
    bn_apply_relu<<<cdiv(N * F1, T), T, 0, stream>>>(z, scale, shift, N * F1, F1);

    // 6) h2 = z @ W2   [N,256]x[256,128]
    {
        int nT64 = F2 / 64;
        int waves = (N / 16) * nT64;
        wmma_gemm_f32<<<cdiv(waves, 8), T, 0, stream>>>(z, W2, h2, N, F1, F2, nT64);
    }

    // 7) out = h2/deg + b2, then scatter edges
    init_self<<<cdiv(N * F2, T), T, 0, stream>>>(h2, dinv, b2, out, N * F2, F2);
    edge_scatter<<<cdiv(E, 8), T, 0, stream>>>(h2, out, src, dst, dinv, E, F2);
}